// GrapherModule_40252433498612
// MI455X (gfx1250) — compile-verified
//
#include <hip/hip_runtime.h>
#include <math.h>

typedef __attribute__((ext_vector_type(2))) float v2f;
typedef __attribute__((ext_vector_type(8))) float v8f;

#define BATCH   4
#define C_IN    192
#define NPTS    1024
#define HDIM    384
#define NHEADS  4
#define KNN     16
#define CHID    1536           // NHEADS * HDIM
#define ROWS    (BATCH * NPTS) // 4096
#define BN_EPS  1e-5f

// ---------------- workspace layout (in floats) ----------------
#define OFF_Y    ((size_t)0)                         // [ROWS, C_IN]        786432
#define OFF_SQ   (OFF_Y    + (size_t)ROWS * C_IN)    // [ROWS]                4096
#define OFF_DIST (OFF_SQ   + (size_t)ROWS)           // [BATCH, NPTS, NPTS] 4194304
#define OFF_H    (OFF_DIST + (size_t)BATCH*NPTS*NPTS)// [ROWS, CHID]        6291456
#define OFF_AS   (OFF_H    + (size_t)ROWS * CHID)    // [ROWS, NHEADS]
#define OFF_AD   (OFF_AS   + (size_t)ROWS * NHEADS)
#define OFF_G    (OFF_AD   + (size_t)ROWS * NHEADS)  // [ROWS, HDIM]
#define OFF_IDX  (OFF_G    + (size_t)ROWS * HDIM)    // [ROWS, KNN] (int)

// f32 WMMA: D(16x16) = A(16x4) * B(4x16) + C, wave32
__device__ __forceinline__ v8f wmma4(v2f a, v2f b, v8f c) {
  return __builtin_amdgcn_wmma_f32_16x16x4_f32(false, a, false, b, (short)0, c,
                                               false, false);
}

// ================= fc1: y = BN1(xT @ W1^T + b1) =================
// x: [B, C, N] (N = H*W), y: [ROWS, C_IN]. Wave tile: 16 rows x 64 cols.
__device__ __forceinline__ void fc1_store(v8f acc, int j, int hi,
                                          const float* __restrict__ b1,
                                          const float* __restrict__ bn1,
                                          float* __restrict__ yb) {
  float bias  = b1[j];
  float scale = bn1[j] / sqrtf(bn1[3 * C_IN + j] + BN_EPS);
  float beta  = bn1[C_IN + j];
  float mean  = bn1[2 * C_IN + j];
#pragma unroll
  for (int r = 0; r < 8; ++r)
    yb[(size_t)(r + 8 * hi) * C_IN + j] = (acc[r] + bias - mean) * scale + beta;
}

__global__ __launch_bounds__(256) void k_fc1(const float* __restrict__ x,
                                             const float* __restrict__ W1,
                                             const float* __restrict__ b1,
                                             const float* __restrict__ bn1,
                                             float* __restrict__ y) {
  const int lane = threadIdx.x & 31;
  const int wave = (blockIdx.x * blockDim.x + threadIdx.x) >> 5; // 0..767
  const int tileJ = wave % 3;              // 3 x 64 cols over 192
  const int tileM = wave / 3;              // 256 x 16 rows over 4096
  const int row0 = tileM * 16;
  const int b    = row0 >> 10;
  const int n0   = row0 & (NPTS - 1);
  const int ml   = lane & 15;
  const int hi   = lane >> 4;
  const int ksel = hi << 1;
  const int j0   = tileJ * 64;
  const float* xb = x + (size_t)b * C_IN * NPTS;

  v8f acc0 = {}, acc1 = {}, acc2 = {}, acc3 = {};
  for (int k = 0; k < C_IN; k += 4) {
    const int kk = k + ksel;
    v2f a;
    a.x = xb[(size_t)kk * NPTS + n0 + ml];
    a.y = xb[(size_t)(kk + 1) * NPTS + n0 + ml];
    v2f bf0 = *(const v2f*)&W1[(size_t)(j0 +  0 + ml) * C_IN + kk];
    v2f bf1 = *(const v2f*)&W1[(size_t)(j0 + 16 + ml) * C_IN + kk];
    v2f bf2 = *(const v2f*)&W1[(size_t)(j0 + 32 + ml) * C_IN + kk];
    v2f bf3 = *(const v2f*)&W1[(size_t)(j0 + 48 + ml) * C_IN + kk];
    acc0 = wmma4(a, bf0, acc0);
    acc1 = wmma4(a, bf1, acc1);
    acc2 = wmma4(a, bf2, acc2);
    acc3 = wmma4(a, bf3, acc3);
  }
  float* yb = y + ((size_t)b * NPTS + n0) * C_IN;
  fc1_store(acc0, j0 +  0 + ml, hi, b1, bn1, yb);
  fc1_store(acc1, j0 + 16 + ml, hi, b1, bn1, yb);
  fc1_store(acc2, j0 + 32 + ml, hi, b1, bn1, yb);
  fc1_store(acc3, j0 + 48 + ml, hi, b1, bn1, yb);
}

// ================= sq[row] = sum_c y[row,c]^2 =================
__global__ __launch_bounds__(256) void k_sq(const float* __restrict__ y,
                                            float* __restrict__ sq) {
  const int lane = threadIdx.x & 31;
  const int row  = (blockIdx.x * blockDim.x + threadIdx.x) >> 5; // 0..4095
  const float* yr = y + (size_t)row * C_IN;
  float s = 0.f;
#pragma unroll
  for (int i = 0; i < C_IN / 32; ++i) {
    float v = yr[lane + i * 32];
    s += v * v;
  }
#pragma unroll
  for (int off = 16; off; off >>= 1) s += __shfl_xor(s, off, 32);
  if (lane == 0) sq[row] = s;
}

// ======= dist[b,n,m] = sq[n] + sq[m] - 2 * (y_n . y_m), via WMMA =======
__global__ __launch_bounds__(256) void k_dist(const float* __restrict__ y,
                                              const float* __restrict__ sq,
                                              float* __restrict__ dist) {
  const int lane = threadIdx.x & 31;
  const int wave = (blockIdx.x * blockDim.x + threadIdx.x) >> 5; // 0..4095
  const int b     = wave >> 10;
  const int rem   = wave & 1023;
  const int n0    = (rem >> 4) * 16;  // 64 row tiles
  const int m0    = (rem & 15) * 64;  // 16 col tiles of 64
  const int ml    = lane & 15;
  const int hi    = lane >> 4;
  const int ksel  = hi << 1;
  const float* yb = y + (size_t)b * NPTS * C_IN;

  v8f acc0 = {}, acc1 = {}, acc2 = {}, acc3 = {};
  for (int k = 0; k < C_IN; k += 4) {
    const int kk = k + ksel;
    v2f a   = *(const v2f*)&yb[(size_t)(n0 + ml) * C_IN + kk];
    v2f bf0 = *(const v2f*)&yb[(size_t)(m0 +  0 + ml) * C_IN + kk];
    v2f bf1 = *(const v2f*)&yb[(size_t)(m0 + 16 + ml) * C_IN + kk];
    v2f bf2 = *(const v2f*)&yb[(size_t)(m0 + 32 + ml) * C_IN + kk];
    v2f bf3 = *(const v2f*)&yb[(size_t)(m0 + 48 + ml) * C_IN + kk];
    acc0 = wmma4(a, bf0, acc0);
    acc1 = wmma4(a, bf1, acc1);
    acc2 = wmma4(a, bf2, acc2);
    acc3 = wmma4(a, bf3, acc3);
  }
  const float* sqb = sq + (size_t)b * NPTS;
  float sqn[8];
#pragma unroll
  for (int r = 0; r < 8; ++r) sqn[r] = sqb[n0 + r + 8 * hi];
  float* db = dist + (size_t)b * NPTS * NPTS;
#pragma unroll
  for (int jj = 0; jj < 4; ++jj) {
    v8f acc = (jj == 0) ? acc0 : (jj == 1) ? acc1 : (jj == 2) ? acc2 : acc3;
    const int mcol = m0 + jj * 16 + ml;
    const float sqm = sqb[mcol];
#pragma unroll
    for (int r = 0; r < 8; ++r) {
      const int n = n0 + r + 8 * hi;
      db[(size_t)n * NPTS + mcol] = sqn[r] + sqm - 2.f * acc[r];
    }
  }
}

// ======= top-16 smallest per row: 16 rounds of wave-wide min-extract =======
__global__ __launch_bounds__(256) void k_topk(const float* __restrict__ dist,
                                              int* __restrict__ idx) {
  const int lane = threadIdx.x & 31;
  const int row  = (blockIdx.x * blockDim.x + threadIdx.x) >> 5; // 0..4095
  const float* dr = dist + (size_t)row * NPTS;
  unsigned long long kv[32];
#pragma unroll
  for (int i = 0; i < 32; ++i) {
    const int m = i * 32 + lane;
    unsigned u = __float_as_uint(dr[m]);
    u = (u & 0x80000000u) ? ~u : (u | 0x80000000u); // order-preserving map
    kv[i] = ((unsigned long long)u << 32) | (unsigned)m;
  }
  int* od = idx + (size_t)row * KNN;
  for (int t = 0; t < KNN; ++t) {
    unsigned long long best = kv[0];
#pragma unroll
    for (int i = 1; i < 32; ++i) best = (kv[i] < best) ? kv[i] : best;
#pragma unroll
    for (int off = 16; off; off >>= 1) {
      unsigned long long o = __shfl_xor(best, off, 32);
      best = (o < best) ? o : best;
    }
    const unsigned mwin = (unsigned)best;
    if (lane == 0) od[t] = (int)mwin;
    if ((int)(mwin & 31u) == lane) {
      const int slot = (int)(mwin >> 5);
#pragma unroll
      for (int i = 0; i < 32; ++i)
        if (i == slot) kv[i] = ~0ull;
    }
  }
}

// ================= h = y @ Wg  (Wg: [C_IN, CHID]) =================
__global__ __launch_bounds__(256) void k_h(const float* __restrict__ y,
                                           const float* __restrict__ Wg,
                                           float* __restrict__ h) {
  const int lane = threadIdx.x & 31;
  const int wave = (blockIdx.x * blockDim.x + threadIdx.x) >> 5; // 0..6143
  const int tileJ = wave % 24;     // 24 x 64 cols over 1536
  const int row0  = (wave / 24) * 16;
  const int ml   = lane & 15;
  const int hi   = lane >> 4;
  const int ksel = hi << 1;
  const int j0   = tileJ * 64;

  v8f acc0 = {}, acc1 = {}, acc2 = {}, acc3 = {};
  for (int k = 0; k < C_IN; k += 4) {
    const int kk = k + ksel;
    v2f a = *(const v2f*)&y[(size_t)(row0 + ml) * C_IN + kk];
    v2f bf0, bf1, bf2, bf3;
    bf0.x = Wg[(size_t)kk * CHID + j0 +  0 + ml];
    bf0.y = Wg[(size_t)(kk + 1) * CHID + j0 +  0 + ml];
    bf1.x = Wg[(size_t)kk * CHID + j0 + 16 + ml];
    bf1.y = Wg[(size_t)(kk + 1) * CHID + j0 + 16 + ml];
    bf2.x = Wg[(size_t)kk * CHID + j0 + 32 + ml];
    bf2.y = Wg[(size_t)(kk + 1) * CHID + j0 + 32 + ml];
    bf3.x = Wg[(size_t)kk * CHID + j0 + 48 + ml];
    bf3.y = Wg[(size_t)(kk + 1) * CHID + j0 + 48 + ml];
    acc0 = wmma4(a, bf0, acc0);
    acc1 = wmma4(a, bf1, acc1);
    acc2 = wmma4(a, bf2, acc2);
    acc3 = wmma4(a, bf3, acc3);
  }
#pragma unroll
  for (int jj = 0; jj < 4; ++jj) {
    v8f acc = (jj == 0) ? acc0 : (jj == 1) ? acc1 : (jj == 2) ? acc2 : acc3;
    const int j = j0 + jj * 16 + ml;
#pragma unroll
    for (int r = 0; r < 8; ++r)
      h[(size_t)(row0 + r + 8 * hi) * CHID + j] = acc[r];
  }
}

// ======= per-(row,head) attention logits: a_src / a_dst =======
__global__ __launch_bounds__(256) void k_attn(const float* __restrict__ h,
                                              const float* __restrict__ att_src,
                                              const float* __restrict__ att_dst,
                                              float* __restrict__ a_src,
                                              float* __restrict__ a_dst) {
  const int lane = threadIdx.x & 31;
  const int wave = (blockIdx.x * blockDim.x + threadIdx.x) >> 5; // 0..16383
  const int head = wave & 3;
  const int row  = wave >> 2;
  const float* hr = h + (size_t)row * CHID + head * HDIM;
  const float* as = att_src + (size_t)head * HDIM;
  const float* ad = att_dst + (size_t)head * HDIM;
  float s = 0.f, d = 0.f;
#pragma unroll
  for (int i = 0; i < HDIM / 32; ++i) {
    const int c = lane + i * 32;
    const float v = hr[c];
    s += v * as[c];
    d += v * ad[c];
  }
#pragma unroll
  for (int off = 16; off; off >>= 1) {
    s += __shfl_xor(s, off, 32);
    d += __shfl_xor(d, off, 32);
  }
  if (lane == 0) {
    a_src[(size_t)row * NHEADS + head] = s;
    a_dst[(size_t)row * NHEADS + head] = d;
  }
}

// ======= GAT aggregation + bias + BN + GELU -> g[row, HDIM] =======
__global__ __launch_bounds__(256) void k_gat(const float* __restrict__ h,
                                             const float* __restrict__ a_src,
                                             const float* __restrict__ a_dst,
                                             const int* __restrict__ idx,
                                             const float* __restrict__ bg,
                                             const float* __restrict__ bng,
                                             float* __restrict__ g) {
  const int lane = threadIdx.x & 31;
  const int row  = (blockIdx.x * blockDim.x + threadIdx.x) >> 5; // 0..4095
  const int b    = row >> 10;
  int nbrRow = 0;
  if (lane < KNN) nbrRow = b * NPTS + idx[(size_t)row * KNN + lane];

  float acc[HDIM / 32];
#pragma unroll
  for (int i = 0; i < HDIM / 32; ++i) acc[i] = 0.f;

  for (int hd = 0; hd < NHEADS; ++hd) {
    const float adst = a_dst[(size_t)row * NHEADS + hd];
    float e = -1e30f;
    if (lane < KNN) {
      const float v = a_src[(size_t)nbrRow * NHEADS + hd] + adst;
      e = (v > 0.f) ? v : 0.2f * v; // leaky_relu(0.2)
    }
    // softmax over the 16 edges (lanes 0..15, width-16 groups)
    float mx = e;
#pragma unroll
    for (int off = 8; off; off >>= 1) mx = fmaxf(mx, __shfl_xor(mx, off, 16));
    float w = (lane < KNN) ? expf(e - mx) : 0.f;
    float sm = w;
#pragma unroll
    for (int off = 8; off; off >>= 1) sm += __shfl_xor(sm, off, 16);
    w = w / sm;

    for (int t = 0; t < KNN; ++t) {
      const float wt = __shfl(w, t, 32);
      const int   nr = __shfl(nbrRow, t, 32);
      const float* hp = h + (size_t)nr * CHID + hd * HDIM;
#pragma unroll
      for (int i = 0; i < HDIM / 32; ++i) acc[i] += wt * hp[lane + i * 32];
    }
  }
#pragma unroll
  for (int i = 0; i < HDIM / 32; ++i) {
    const int d = lane + i * 32;
    float v = acc[i] * 0.25f + bg[d]; // mean over 4 heads + bias
    const float scale = bng[d] / sqrtf(bng[3 * HDIM + d] + BN_EPS);
    v = (v - bng[2 * HDIM + d]) * scale + bng[HDIM + d];
    v = 0.5f * v * (1.f + erff(v * 0.70710678118654752f)); // exact GELU
    g[(size_t)row * HDIM + d] = v;
  }
}

// ======= fc2: out = BN2(g @ W2^T + b2) + x, stored as [B,C,N] =======
__global__ __launch_bounds__(256) void k_fc2(const float* __restrict__ g,
                                             const float* __restrict__ W2,
                                             const float* __restrict__ b2,
                                             const float* __restrict__ bn2,
                                             const float* __restrict__ x,
                                             float* __restrict__ out) {
  const int lane = threadIdx.x & 31;
  const int wave = (blockIdx.x * blockDim.x + threadIdx.x) >> 5; // 0..767
  const int tileJ = wave % 3;
  const int row0  = (wave / 3) * 16;
  const int b     = row0 >> 10;
  const int n0    = row0 & (NPTS - 1);
  const int ml    = lane & 15;
  const int hi    = lane >> 4;
  const int ksel  = hi << 1;
  const int j0    = tileJ * 64;

  v8f acc0 = {}, acc1 = {}, acc2 = {}, acc3 = {};
  for (int k = 0; k < HDIM; k += 4) {
    const int kk = k + ksel;
    v2f a   = *(const v2f*)&g[(size_t)(row0 + ml) * HDIM + kk];
    v2f bf0 = *(const v2f*)&W2[(size_t)(j0 +  0 + ml) * HDIM + kk];
    v2f bf1 = *(const v2f*)&W2[(size_t)(j0 + 16 + ml) * HDIM + kk];
    v2f bf2 = *(const v2f*)&W2[(size_t)(j0 + 32 + ml) * HDIM + kk];
    v2f bf3 = *(const v2f*)&W2[(size_t)(j0 + 48 + ml) * HDIM + kk];
    acc0 = wmma4(a, bf0, acc0);
    acc1 = wmma4(a, bf1, acc1);
    acc2 = wmma4(a, bf2, acc2);
    acc3 = wmma4(a, bf3, acc3);
  }
  const float* xb = x + (size_t)b * C_IN * NPTS;
  float* ob = out + (size_t)b * C_IN * NPTS;
#pragma unroll
  for (int jj = 0; jj < 4; ++jj) {
    v8f acc = (jj == 0) ? acc0 : (jj == 1) ? acc1 : (jj == 2) ? acc2 : acc3;
    const int j = j0 + jj * 16 + ml;
    const float bias  = b2[j];
    const float scale = bn2[j] / sqrtf(bn2[3 * C_IN + j] + BN_EPS);
    const float beta  = bn2[C_IN + j];
    const float mean  = bn2[2 * C_IN + j];
#pragma unroll
    for (int r = 0; r < 8; ++r) {
      const int n = n0 + r + 8 * hi;
      const float v = (acc[r] + bias - mean) * scale + beta;
      ob[(size_t)j * NPTS + n] = v + xb[(size_t)j * NPTS + n];
    }
  }
}

extern "C" void kernel_launch(void* const* d_in, const int* in_sizes, int n_in,
                              void* d_out, int out_size, void* d_ws, size_t ws_size,
                              hipStream_t stream) {
  (void)in_sizes; (void)n_in; (void)out_size; (void)ws_size;
  const float* x       = (const float*)d_in[0];
  const float* W1      = (const float*)d_in[1];
  const float* b1      = (const float*)d_in[2];
  const float* bn1     = (const float*)d_in[3];
  const float* Wg      = (const float*)d_in[4];
  const float* att_src = (const float*)d_in[5];
  const float* att_dst = (const float*)d_in[6];
  const float* bg      = (const float*)d_in[7];
  const float* bng     = (const float*)d_in[8];
  const float* W2      = (const float*)d_in[9];
  const float* b2      = (const float*)d_in[10];
  const float* bn2     = (const float*)d_in[11];
  float* out = (float*)d_out;

  float* ws    = (float*)d_ws;
  float* y     = ws + OFF_Y;
  float* sq    = ws + OFF_SQ;
  float* dist  = ws + OFF_DIST;
  float* h     = ws + OFF_H;
  float* a_src = ws + OFF_AS;
  float* a_dst = ws + OFF_AD;
  float* g     = ws + OFF_G;
  int*   idx   = (int*)(ws + OFF_IDX);

  k_fc1 <<<  96, 256, 0, stream>>>(x, W1, b1, bn1, y);          // 768 waves
  k_sq  <<< 512, 256, 0, stream>>>(y, sq);                      // 4096 waves
  k_dist<<< 512, 256, 0, stream>>>(y, sq, dist);                // 4096 waves
  k_topk<<< 512, 256, 0, stream>>>(dist, idx);                  // 4096 waves
  k_h   <<< 768, 256, 0, stream>>>(y, Wg, h);                   // 6144 waves
  k_attn<<<2048, 256, 0, stream>>>(h, att_src, att_dst, a_src, a_dst);
  k_gat <<< 512, 256, 0, stream>>>(h, a_src, a_dst, idx, bg, bng, g);
  k_fc2 <<<  96, 256, 0, stream>>>(g, W2, b2, bn2, x, out);
}